// PredictiveAlignmentAttention_32160715113134
// MI455X (gfx1250) — compile-verified
//
#include <hip/hip_runtime.h>
#include <hip/hip_bf16.h>
#include <math.h>

typedef __attribute__((ext_vector_type(2))) float v2f;
typedef __attribute__((ext_vector_type(4))) float v4f;
typedef __attribute__((ext_vector_type(8))) float v8f;

#define B_DIM 128
#define H_DIM 1024
#define FH 24
#define FW 48
#define SX2x2 32.0f   // 2*SX*SX
#define SY2x2 32.0f   // 2*SY*SY

// ---------------------------------------------------------------------------
// WMMA f32 GEMM:  out[M,N] = epi( X[M,K] @ W[N,K]^T + bias[N] )
// One wave computes one 16x16 tile via V_WMMA_F32_16X16X4_F32.
// EPI: 0 = tanh, 1 = relu.
// Launch: blockDim = 256 (8 waves), gridDim.x = (M/16)*(N/16)/8 exactly.
// ---------------------------------------------------------------------------
template <int EPI>
__global__ void paa_gemm_wmma(const float* __restrict__ X,
                              const float* __restrict__ W,
                              const float* __restrict__ bias,
                              float* __restrict__ out,
                              int M, int N, int K) {
    const int lane = threadIdx.x & 31;
    const int wave = threadIdx.x >> 5;
    const int tile = blockIdx.x * 8 + wave;
    const int ntN  = N >> 4;
    const int m0   = (tile / ntN) << 4;
    const int n0   = (tile % ntN) << 4;
    const int i    = lane & 15;   // A row / B col within tile
    const int g    = lane >> 4;   // K half-group

    // lane reads K elements {kk+2g, kk+2g+1} of A row (m0+i) and W row (n0+i)
    const float* arow = X + (size_t)(m0 + i) * K + 2 * g;
    const float* brow = W + (size_t)(n0 + i) * K + 2 * g;

    v8f acc = {};
#pragma unroll 4
    for (int kk = 0; kk < K; kk += 4) {
        v2f a = *(const v2f*)(arow + kk);   // 8B-aligned (row 16B, +2g = 8B)
        v2f b = *(const v2f*)(brow + kk);
        acc = __builtin_amdgcn_wmma_f32_16x16x4_f32(
            /*neg_a=*/false, a, /*neg_b=*/false, b,
            /*c_mod=*/(short)0, acc, /*reuse_a=*/false, /*reuse_b=*/false);
    }

    const float bn = bias[n0 + i];
#pragma unroll
    for (int v = 0; v < 8; ++v) {
        const int m = m0 + v + 8 * g;          // C/D layout: VGPR v, group g -> M = v + 8g
        float x = acc[v] + bn;
        x = (EPI == 0) ? tanhf(x) : fmaxf(x, 0.0f);
        out[(size_t)m * N + (n0 + i)] = x;
    }
}

// ---------------------------------------------------------------------------
// algn[b, j] = sigmoid( A1[b,:] . W2[j,:] + b2[j] ),  b in [0,128), j in {0,1}
// One block of 256 threads.
// ---------------------------------------------------------------------------
__global__ void paa_algn(const float* __restrict__ A1,
                         const float* __restrict__ W2,
                         const float* __restrict__ b2,
                         float* __restrict__ algn) {
    const int t = threadIdx.x;
    const int b = t >> 1;
    const int j = t & 1;
    const float* xr = A1 + (size_t)b * (2 * H_DIM);
    const float* wr = W2 + (size_t)j * (2 * H_DIM);
    float s = b2[j];
#pragma unroll 8
    for (int k = 0; k < 2 * H_DIM; ++k) s = fmaf(xr[k], wr[k], s);
    algn[b * 2 + j] = 1.0f / (1.0f + expf(-s));
}

// ---------------------------------------------------------------------------
// Separable normalized Gaussian weights per batch:
//   wxn[b,c] = exp(-(c-x0)^2/32) / sum_c(...),  wyn[b,r] likewise.
// One block of 128 threads, one thread per batch.
// ---------------------------------------------------------------------------
__global__ void paa_weights(const float* __restrict__ algn,
                            float* __restrict__ wxn,
                            float* __restrict__ wyn) {
    const int b = threadIdx.x;
    const float x0 = algn[2 * b + 0] * (float)FW;
    const float y0 = algn[2 * b + 1] * (float)FH;

    float sx = 0.0f;
    for (int c = 0; c < FW; ++c) { float d = (float)c - x0; sx += __expf(-d * d / SX2x2); }
    float ix = 1.0f / sx;
    for (int c = 0; c < FW; ++c) { float d = (float)c - x0; wxn[b * FW + c] = __expf(-d * d / SX2x2) * ix; }

    float sy = 0.0f;
    for (int r = 0; r < FH; ++r) { float d = (float)r - y0; sy += __expf(-d * d / SY2x2); }
    float iy = 1.0f / sy;
    for (int r = 0; r < FH; ++r) { float d = (float)r - y0; wyn[b * FH + r] = __expf(-d * d / SY2x2) * iy; }
}

// ---------------------------------------------------------------------------
// Streaming einsum partials (the 604 MB kernel):
//   partial[b, r, h] = wyn[b,r] * sum_c wxn[b,c] * V[r, c, b, h]
// grid = (B, FH), block = 256 threads, each thread owns 4 consecutive h
// -> every c-iteration streams one contiguous 4 KB slab per block (coalesced b128).
// ---------------------------------------------------------------------------
__global__ void paa_attn_partial(const float* __restrict__ V,
                                 const float* __restrict__ wxn,
                                 const float* __restrict__ wyn,
                                 float* __restrict__ partial) {
    const int b = blockIdx.x;
    const int r = blockIdx.y;
    const int h4 = threadIdx.x * 4;

    __shared__ float swx[FW];
    if (threadIdx.x < FW) swx[threadIdx.x] = wxn[b * FW + threadIdx.x];
    __syncthreads();

    const float* vbase = V + ((size_t)(r * FW) * B_DIM + b) * H_DIM + h4;
    v4f acc = {};
#pragma unroll 4
    for (int c = 0; c < FW; ++c) {
        v4f v = *(const v4f*)(vbase + (size_t)c * (B_DIM * H_DIM));
        acc += swx[c] * v;
    }
    const float scale = wyn[b * FH + r];
    *(v4f*)(partial + ((size_t)(b * FH + r) * H_DIM) + h4) = scale * acc;
}

// ---------------------------------------------------------------------------
// Reduce partials over r and build Xcat = [inp | attn]  (128 x 2048).
// 512 blocks x 256 threads = 131072 threads, one per (b,h).
// ---------------------------------------------------------------------------
__global__ void paa_concat_reduce(const float* __restrict__ inp,
                                  const float* __restrict__ partial,
                                  float* __restrict__ Xcat) {
    const int flat = blockIdx.x * 256 + threadIdx.x;
    const int b = flat >> 10;
    const int h = flat & (H_DIM - 1);
    float s = 0.0f;
#pragma unroll
    for (int r = 0; r < FH; ++r)
        s += partial[((size_t)(b * FH + r) << 10) + h];
    Xcat[(size_t)b * (2 * H_DIM) + h] = inp[flat];
    Xcat[(size_t)b * (2 * H_DIM) + H_DIM + h] = s;
}

// ---------------------------------------------------------------------------
extern "C" void kernel_launch(void* const* d_in, const int* in_sizes, int n_in,
                              void* d_out, int out_size, void* d_ws, size_t ws_size,
                              hipStream_t stream) {
    const float* inp = (const float*)d_in[0];
    const float* hid = (const float*)d_in[1];
    const float* V   = (const float*)d_in[2];
    const float* W1  = (const float*)d_in[3];
    const float* b1  = (const float*)d_in[4];
    const float* W2  = (const float*)d_in[5];
    const float* b2  = (const float*)d_in[6];
    const float* Wc  = (const float*)d_in[7];
    const float* bc  = (const float*)d_in[8];
    float* out = (float*)d_out;

    // workspace layout (floats)
    float* ws      = (float*)d_ws;
    float* A1      = ws;                              // 128*2048
    float* algn    = A1 + B_DIM * 2 * H_DIM;          // 256
    float* wxn     = algn + 2 * B_DIM;                // 128*48
    float* wyn     = wxn + B_DIM * FW;                // 128*24
    float* partial = wyn + B_DIM * FH;                // 128*24*1024
    float* Xcat    = partial + (size_t)B_DIM * FH * H_DIM; // 128*2048

    // 1) A1 = tanh(hid @ W1^T + b1) : tiles = (128/16)*(2048/16) = 1024 -> 128 blocks
    paa_gemm_wmma<0><<<128, 256, 0, stream>>>(hid, W1, b1, A1, B_DIM, 2 * H_DIM, H_DIM);

    // 2) algn = sigmoid(A1 @ W2^T + b2)
    paa_algn<<<1, 256, 0, stream>>>(A1, W2, b2, algn);

    // 3) separable normalized Gaussian weights
    paa_weights<<<1, B_DIM, 0, stream>>>(algn, wxn, wyn);

    // 4) streaming einsum partials over V (bandwidth-bound: 604 MB)
    paa_attn_partial<<<dim3(B_DIM, FH), 256, 0, stream>>>(V, wxn, wyn, partial);

    // 5) reduce over r + concat with inp
    paa_concat_reduce<<<512, 256, 0, stream>>>(inp, partial, Xcat);

    // 6) ctx = relu(Xcat @ Wc^T + bc) : tiles = (128/16)*(1024/16) = 512 -> 64 blocks
    paa_gemm_wmma<1><<<64, 256, 0, stream>>>(Xcat, Wc, bc, out, B_DIM, H_DIM, 2 * H_DIM);
}